// drqssm_64407329570984
// MI455X (gfx1250) — compile-verified
//
#include <hip/hip_runtime.h>

// ---- problem constants -----------------------------------------------------
#define BB   4
#define DD   192
#define NN   16
#define RR   6
#define LL   256
#define LX   255
#define KTOT 38          // R + 2N columns of x_dbl

typedef float v2f __attribute__((ext_vector_type(2)));
typedef float v8f __attribute__((ext_vector_type(8)));

// ---- workspace layout (float offsets) -------------------------------------
#define WS_XDBL 0                                 // B*L*38      = 38912
#define WS_DT   (WS_XDBL + BB*LL*KTOT)            // B*D*L       = 196608
#define WS_YS   (WS_DT   + BB*DD*LL)              // B*D*L       = 196608
#define WS_S    (WS_YS   + BB*DD*LL)              // B*D*L*N     = 3145728
// total = 3,577,856 floats ~= 14.3 MB

// Fast exp for arguments known to be <= 0 (or tiny positive): native
// v_exp_f32 path, no range-fixup code. Underflow flushes to 0 == exact answer.
__device__ __forceinline__ float fexp(float x) { return __expf(x); }

// ---------------------------------------------------------------------------
// K1: x_dbl[b,l,k] = sum_d xf[b,d,l] * W[k,d]   (xf = concat(f, x) along L)
// ---------------------------------------------------------------------------
__global__ __launch_bounds__(256)
void k_proj(const float* __restrict__ x, const float* __restrict__ f,
            const float* __restrict__ W, float* __restrict__ ws)
{
    int tid = blockIdx.x * blockDim.x + threadIdx.x;
    if (tid >= BB * LL * KTOT) return;
    int k = tid % KTOT;
    int l = (tid / KTOT) % LL;
    int b = tid / (KTOT * LL);
    const float* wrow = W + k * DD;
    float acc = 0.f;
    if (l == 0) {
        const float* fb = f + b * DD;                       // f[b,d,0]
        for (int d = 0; d < DD; ++d) acc += fb[d] * wrow[d];
    } else {
        const float* xb = x + (size_t)b * DD * LX + (l - 1); // x[b,d,l-1]
        for (int d = 0; d < DD; ++d) acc += xb[(size_t)d * LX] * wrow[d];
    }
    ws[WS_XDBL + (b * LL + l) * KTOT + k] = acc;
}

// ---------------------------------------------------------------------------
// K2: dt[b,d,l] = softplus( sum_r x_dbl[b,l,r]*dtw[d,r] + bias[d] )
//     (precise expf/log1pf: dt feeds the exponents, keep accuracy here)
// ---------------------------------------------------------------------------
__global__ __launch_bounds__(256)
void k_dt(const float* __restrict__ dtw, const float* __restrict__ dtb,
          float* __restrict__ ws)
{
    int tid = blockIdx.x * blockDim.x + threadIdx.x;
    if (tid >= BB * DD * LL) return;
    int l = tid % LL;
    int d = (tid / LL) % DD;
    int b = tid / (LL * DD);
    const float* xd = ws + WS_XDBL + (b * LL + l) * KTOT;
    float v = dtb[d];
    #pragma unroll
    for (int r = 0; r < RR; ++r) v += xd[r] * dtw[d * RR + r];
    float sp = (v > 30.f) ? v : log1pf(expf(v));
    ws[WS_DT + (b * DD + d) * LL + l] = sp;
}

// ---------------------------------------------------------------------------
// K3: sequential scan. One 16-lane half-wave per (b,d) channel.
//     lanes 0..15 -> channel 2w, lanes 16..31 -> channel 2w+1; lane%16 = n.
//     Produces S[b,d,l,n] (cumsum of log_dA) and ys_raw[b,d,l].
// ---------------------------------------------------------------------------
__global__ __launch_bounds__(256)
void k_scan(const float* __restrict__ x, const float* __restrict__ f,
            const float* __restrict__ A_logs, const float* __restrict__ Ds,
            float* __restrict__ ws)
{
    int lane = threadIdx.x & 31;
    int wv   = (blockIdx.x * blockDim.x + threadIdx.x) >> 5;
    int hi   = lane >> 4;
    int n    = lane & 15;
    int ch   = wv * 2 + hi;              // 0 .. B*D-1
    int b    = ch / DD;
    int d    = ch % DD;

    float An  = -expf(A_logs[d * NN + n]);   // precise
    float Dsd = Ds[d];
    const float* dtp  = ws + WS_DT   + (size_t)(b * DD + d) * LL;
    const float* xdbl = ws + WS_XDBL + (size_t)b * LL * KTOT;
    float* Sp  = ws + WS_S  + (size_t)(b * DD + d) * LL * NN;
    float* ysp = ws + WS_YS + (size_t)(b * DD + d) * LL;

    float h = 0.f, srun = 0.f;
    for (int l = 0; l < LL; ++l) {
        float dtv = dtp[l];
        float Bsv = xdbl[l * KTOT + RR + n];
        float Csv = xdbl[l * KTOT + RR + NN + n];
        float u   = (l == 0) ? f[b * DD + d]
                             : x[(size_t)(b * DD + d) * LX + (l - 1)];
        float lda = dtv * An;                // <= 0 always
        srun += lda;
        Sp[l * NN + n] = srun;
        h = fexp(lda) * h + dtv * Bsv * u;   // native exp, arg <= 0
        float p = h * Csv;                   // reduce over n (16 lanes)
        p += __shfl_xor(p, 1);
        p += __shfl_xor(p, 2);
        p += __shfl_xor(p, 4);
        p += __shfl_xor(p, 8);
        if (n == 0) ysp[l] = p + Dsd * u;
    }
}

// ---------------------------------------------------------------------------
// K4: LayerNorm over d for l = 1..255; writes y output (b, l-1, d).
// ---------------------------------------------------------------------------
__global__ __launch_bounds__(256)
void k_ln(const float* __restrict__ lnw, const float* __restrict__ lnb,
          const float* __restrict__ ws, float* __restrict__ yout)
{
    int bl = blockIdx.x;                  // 0 .. B*LX-1
    int b  = bl / LX;
    int l  = bl % LX + 1;                 // 1..255
    int t  = threadIdx.x;                 // 256 threads
    __shared__ float red[256];

    float v = (t < DD) ? ws[WS_YS + (size_t)(b * DD + t) * LL + l] : 0.f;
    red[t] = v;
    __syncthreads();
    for (int s = 128; s > 0; s >>= 1) {
        if (t < s) red[t] += red[t + s];
        __syncthreads();
    }
    float mu = red[0] * (1.f / DD);
    __syncthreads();
    float dv = (t < DD) ? (v - mu) : 0.f;
    red[t] = dv * dv;
    __syncthreads();
    for (int s = 128; s > 0; s >>= 1) {
        if (t < s) red[t] += red[t + s];
        __syncthreads();
    }
    float var = red[0] * (1.f / DD);
    float rs  = rsqrtf(var + 1e-5f);
    if (t < DD)
        yout[(size_t)(b * LX + (l - 1)) * DD + t] = (v - mu) * rs * lnw[t] + lnb[t];
}

// ---------------------------------------------------------------------------
// K5: attn. One wave per 16x16 tile of the (L x L) matrix per (b,d).
//   ti < tj : zeros (float4 stores). ti == tj : direct clamped form.
//   ti > tj : anchored factorization, 4 chained V_WMMA_F32_16X16X4_F32.
//   All exp arguments are <= 0 by construction -> native __expf is exact
//   enough (underflow->0 is the true value; no overflow side exists).
// ---------------------------------------------------------------------------
__global__ __launch_bounds__(256)
void k_attn(const float* __restrict__ ws, float* __restrict__ attn)
{
    int lane = threadIdx.x & 31;
    // wave-uniform job id (readfirstlane => scalar branches, EXEC all-ones)
    int wvInBlk = __builtin_amdgcn_readfirstlane(threadIdx.x >> 5);
    int wv = blockIdx.x * 8 + wvInBlk;    // 0 .. B*D*256-1
    int tj = wv & 15;
    int ti = (wv >> 4) & 15;
    int bd = wv >> 8;                      // 0 .. B*D-1
    int b  = bd / DD;
    int row = lane & 15;
    int hi  = lane >> 4;
    int I0 = ti * 16, J0 = tj * 16;

    float* out = attn + ((size_t)bd * LL + I0) * LL + J0;

    if (ti < tj) {                         // above diagonal: zero fill
        // 16 rows x 16 cols: each lane stores two float4s (64B-aligned rows)
        int zr = lane >> 1;                // row 0..15
        int zc = (lane & 1) * 8;           // col 0 or 8
        float4 z4 = make_float4(0.f, 0.f, 0.f, 0.f);
        *(float4*)(out + zr * LL + zc)     = z4;
        *(float4*)(out + zr * LL + zc + 4) = z4;
        return;
    }

    const float* Sbd  = ws + WS_S    + (size_t)bd * LL * NN;
    const float* xdbl = ws + WS_XDBL + (size_t)b * LL * KTOT;
    const float* dtp  = ws + WS_DT   + (size_t)bd * LL;

    if (ti == tj) {                        // diagonal tile: direct, clamped
        int i = I0 + row;
        float Si[NN], Ci[NN];
        #pragma unroll
        for (int n = 0; n < NN; ++n) {
            Si[n] = Sbd[i * NN + n];
            Ci[n] = xdbl[i * KTOT + RR + NN + n];
        }
        #pragma unroll
        for (int t8 = 0; t8 < 8; ++t8) {
            int jl = hi * 8 + t8;
            int j  = J0 + jl;
            float acc = 0.f;
            if (i >= j) {
                float dtj = dtp[j];
                #pragma unroll
                for (int n = 0; n < NN; ++n) {
                    float Sj  = Sbd[j * NN + n];
                    float Bsj = xdbl[j * KTOT + RR + n];
                    acc += Ci[n] * fexp(fminf(Si[n] - Sj, 0.f)) * dtj * Bsj;
                }
            }
            out[row * LL + jl] = acc;
        }
        return;
    }

    // strictly-lower tile: attn = P @ Q^T via V_WMMA_F32_16X16X4_F32 x4.
    // Per ISA f32 layouts: A 16x4 -> lane(row,hi) holds K = {4c+2hi, 4c+2hi+1};
    // B 4x16 symmetric; C/D 16x16 -> VGPR v at lane = (M = v+8hi, N = row).
    int i = I0 + row;
    int j = J0 + row;
    float dtj = dtp[j];
    float Pa[4], Pb[4], Qa[4], Qb[4];
    #pragma unroll
    for (int c = 0; c < 4; ++c) {
        int n0 = 4 * c + 2 * hi;
        float r0 = Sbd[I0 * NN + n0];
        float r1 = Sbd[I0 * NN + n0 + 1];
        Pa[c] = xdbl[i * KTOT + RR + NN + n0    ] * fexp(Sbd[i * NN + n0    ] - r0);
        Pb[c] = xdbl[i * KTOT + RR + NN + n0 + 1] * fexp(Sbd[i * NN + n0 + 1] - r1);
        Qa[c] = dtj * xdbl[j * KTOT + RR + n0    ] * fexp(r0 - Sbd[j * NN + n0    ]);
        Qb[c] = dtj * xdbl[j * KTOT + RR + n0 + 1] * fexp(r1 - Sbd[j * NN + n0 + 1]);
    }
    v8f acc = {};
    #pragma unroll
    for (int c = 0; c < 4; ++c) {
        v2f a  = {Pa[c], Pb[c]};
        v2f bv = {Qa[c], Qb[c]};
        acc = __builtin_amdgcn_wmma_f32_16x16x4_f32(
                  /*neg_a=*/false, a, /*neg_b=*/false, bv,
                  /*c_mod=*/(short)0, acc, /*reuse_a=*/false, /*reuse_b=*/false);
    }
    #pragma unroll
    for (int v = 0; v < 8; ++v)
        out[(v + 8 * hi) * LL + row] = acc[v];
}

// ---------------------------------------------------------------------------
extern "C" void kernel_launch(void* const* d_in, const int* in_sizes, int n_in,
                              void* d_out, int out_size, void* d_ws, size_t ws_size,
                              hipStream_t stream)
{
    const float* x    = (const float*)d_in[0];
    const float* f    = (const float*)d_in[1];
    const float* xpw  = (const float*)d_in[2];
    const float* dtw  = (const float*)d_in[3];
    const float* dtb  = (const float*)d_in[4];
    const float* Alog = (const float*)d_in[5];
    const float* Ds   = (const float*)d_in[6];
    const float* lnw  = (const float*)d_in[7];
    const float* lnb  = (const float*)d_in[8];

    float* ws   = (float*)d_ws;
    float* yout = (float*)d_out;                       // B*LX*DD floats
    float* attn = yout + (size_t)BB * LX * DD;         // B*D*L*L floats

    k_proj<<<(BB * LL * KTOT + 255) / 256, 256, 0, stream>>>(x, f, xpw, ws);
    k_dt  <<<(BB * DD * LL + 255) / 256, 256, 0, stream>>>(dtw, dtb, ws);
    k_scan<<<(BB * DD) / 16, 256, 0, stream>>>(x, f, Alog, Ds, ws);
    k_ln  <<<BB * LX, 256, 0, stream>>>(lnw, lnb, ws, yout);
    k_attn<<<(BB * DD * 256) / 8, 256, 0, stream>>>(ws, attn);
}